// MaxPlusMP_13099650252927
// MI455X (gfx1250) — compile-verified
//
#include <hip/hip_runtime.h>
#include <stdint.h>

// ---------------------------------------------------------------------------
// Max-plus matmul: y[n,o] = max_i (x[n,i] + a[o,i]) + bias[o]
//   x: [4096, 256] f32, a: [256, 256] f32, bias: [256] f32, y: [4096, 256] f32
//
// CDNA5 strategy: no WMMA path exists for the (max,+) semiring, so compute is
// VALU v_add_f32 + v_max_num_f32 (VOPD dual-issued) with 4x4 register
// blocking. Data movement uses the Tensor Data Mover (tensor_load_to_lds,
// TENSORcnt) with hardware LDS padding (row stride 33 floats) for
// bank-conflict-free reads, double buffered so DMA overlaps compute.
// ---------------------------------------------------------------------------

typedef __attribute__((ext_vector_type(4))) unsigned int v4u;
typedef __attribute__((ext_vector_type(8))) int          v8i;
typedef __attribute__((ext_vector_type(4))) int          v4i;

#define N_ROWS 4096
#define IN_DIM 256
#define OUT_DIM 256

#define BM 64
#define BN 64
#define BK 32
#define LDS_STRIDE 33                 // floats per row after TDM pad (32 + 1)
#define TILE_FLOATS (BM * LDS_STRIDE) // 2112 floats per tile
#define TILE_BYTES  (TILE_FLOATS * 4) // 8448 bytes
#define BUF_FLOATS  (2 * TILE_FLOATS) // x-tile + a-tile per buffer
#define NK (IN_DIM / BK)              // 8 K-steps

// Issue one 2D TDM load: tile_d1 rows x BK cols of f32 from a row-major
// tensor (row length/stride IN_DIM) starting at gptr, into LDS at byte
// offset lds_byte_off, padding 1 dword after every 32 dwords.
__device__ __forceinline__ void tdm_load_tile_f32(unsigned lds_byte_off,
                                                  const float* gptr,
                                                  unsigned tensor_d1,
                                                  unsigned tile_d1) {
  uint64_t ga = (uint64_t)(uintptr_t)gptr;

  // D# group 0 (128b): count=1 | lds_addr | global_addr[56:0] | type=2
  v4u g0;
  g0.x = 1u;                                       // count=1 (valid user desc)
  g0.y = lds_byte_off;                             // LDS byte address
  g0.z = (unsigned)(ga & 0xFFFFFFFFu);             // global_addr[31:0]
  g0.w = (unsigned)((ga >> 32) & 0x01FFFFFFu)      // global_addr[56:32]
       | (2u << 30);                               // type=2 ("image")

  // D# group 1 (256b)
  v8i g1;
  g1.s0 = (int)((2u << 16)                         // data_size = 4 bytes
              | (1u << 20)                         // pad_enable
              | (4u << 22)                         // pad_interval: 32 dwords
              | (0u << 25));                       // pad_amount: 1 dword
  g1.s1 = (int)((unsigned)IN_DIM << 16);           // tensor_dim0[15:0]
  g1.s2 = (int)(((unsigned)IN_DIM >> 16) | (tensor_d1 << 16));
  g1.s3 = (int)((tensor_d1 >> 16) | ((unsigned)BK << 16)); // tile_dim0 = BK
  g1.s4 = (int)(tile_d1 & 0xFFFFu);                // tile_dim1 (tile_dim2=0)
  g1.s5 = (int)IN_DIM;                             // tensor_dim0_stride
  g1.s6 = 0;
  g1.s7 = 0;

  v4i gz4 = {0, 0, 0, 0};                          // groups 2/3 unused (2D)
  v8i gz8 = {0, 0, 0, 0, 0, 0, 0, 0};              // trailing group (clang-23)
  __builtin_amdgcn_tensor_load_to_lds(g0, g1, gz4, gz4, gz8, 0);
}

__global__ __launch_bounds__(256) void maxplus_tdm_kernel(
    const float* __restrict__ x, const float* __restrict__ a,
    const float* __restrict__ bias, float* __restrict__ y) {
  // Double-buffered LDS: [buf][x-tile | a-tile], row stride 33 floats.
  __shared__ float smem[2 * BUF_FLOATS];           // 33792 bytes

  const int tid = threadIdx.x;
  const int tx  = tid & 15;                        // output-column group
  const int ty  = tid >> 4;                        // output-row group
  const int rowBase = blockIdx.x * BM;
  const int colBase = blockIdx.y * BN;

  float acc[4][4];
#pragma unroll
  for (int m = 0; m < 4; ++m)
#pragma unroll
    for (int n = 0; n < 4; ++n) acc[m][n] = -__builtin_inff();

  float bv[4];
#pragma unroll
  for (int n = 0; n < 4; ++n) bv[n] = bias[colBase + tx * 4 + n];

  // Prologue: wave 0 issues DMA for buffer 0 (k0 = 0).
  if (tid < 32) {
    tdm_load_tile_f32(0u, x + (size_t)rowBase * IN_DIM, N_ROWS, BM);
    tdm_load_tile_f32(TILE_BYTES, a + (size_t)colBase * IN_DIM, OUT_DIM, BN);
  }

#pragma unroll 1
  for (int t = 0; t < NK; ++t) {
    const int b = t & 1;

    if (tid < 32) __builtin_amdgcn_s_wait_tensorcnt(0); // buffer b complete
    __syncthreads();  // publish buffer b; protects buffer b^1 for overwrite

    // The TDM writes LDS behind the compiler's back (the descriptor carries
    // only an integer LDS offset). Escape smem's address and clobber memory
    // so the ds_loads below are not folded to undef.
    asm volatile("" : : "v"(&smem[0]) : "memory");

    // Prefetch next K-slice into the other buffer while we compute.
    if (tid < 32 && (t + 1) < NK) {
      const int k0 = (t + 1) * BK;
      const unsigned boff = (unsigned)((b ^ 1) * (BUF_FLOATS * 4));
      tdm_load_tile_f32(boff, x + (size_t)rowBase * IN_DIM + k0, N_ROWS, BM);
      tdm_load_tile_f32(boff + TILE_BYTES, a + (size_t)colBase * IN_DIM + k0,
                        OUT_DIM, BN);
    }

    const float* xs = &smem[b * BUF_FLOATS];
    const float* as = xs + TILE_FLOATS;

#pragma unroll
    for (int k = 0; k < BK; ++k) {
      float xr[4], ar[4];
#pragma unroll
      for (int m = 0; m < 4; ++m)
        xr[m] = xs[(ty * 4 + m) * LDS_STRIDE + k];  // broadcast across tx
#pragma unroll
      for (int n = 0; n < 4; ++n)
        ar[n] = as[(tx * 4 + n) * LDS_STRIDE + k];  // 16 distinct banks
#pragma unroll
      for (int m = 0; m < 4; ++m)
#pragma unroll
        for (int n = 0; n < 4; ++n)
          acc[m][n] = fmaxf(acc[m][n], xr[m] + ar[n]);
    }
    // Single barrier per iteration (top of loop) is sufficient: the barrier
    // at step t+1 both publishes buffer t+1 and guarantees everyone finished
    // reading buffer t before the DMA issued at step t+1 (slice t+2) overwrites it.
  }

#pragma unroll
  for (int m = 0; m < 4; ++m) {
    const int row = rowBase + ty * 4 + m;
    float4 v;
    v.x = acc[m][0] + bv[0];
    v.y = acc[m][1] + bv[1];
    v.z = acc[m][2] + bv[2];
    v.w = acc[m][3] + bv[3];
    *(float4*)&y[(size_t)row * OUT_DIM + colBase + tx * 4] = v;
  }
}

extern "C" void kernel_launch(void* const* d_in, const int* in_sizes, int n_in,
                              void* d_out, int out_size, void* d_ws,
                              size_t ws_size, hipStream_t stream) {
  const float* x    = (const float*)d_in[0];
  const float* a    = (const float*)d_in[1];
  const float* bias = (const float*)d_in[2];
  float* y          = (float*)d_out;

  dim3 grid(N_ROWS / BM, OUT_DIM / BN);   // 64 x 4 blocks
  maxplus_tdm_kernel<<<grid, 256, 0, stream>>>(x, a, bias, y);
}